// Disen_GCN_84911503442544
// MI455X (gfx1250) — compile-verified
//
#include <hip/hip_runtime.h>
#include <hip/hip_bf16.h>

typedef __attribute__((ext_vector_type(2))) float v2f;
typedef __attribute__((ext_vector_type(8))) float v8f;

#define N_NODES 4096
#define C_CH    8
#define IN_DIM  256
#define D_DIM   64
#define OUT_DIM 128
#define ITERS   4

// ---------------------------------------------------------------------------
// Kernel 1: z[c,n,d] = l2norm_d( features[n,:] @ W[c,:,d] + b[c,d] )
// One wave per (channel, 16-row tile). F32 WMMA 16x16x4, K=256 in 64 steps.
// A 16x4 layout: lane L holds row M=L&15, K = 2*(L>>4)+{0,1} in {a.x,a.y}.
// B 4x16 layout: lane L holds col N=L&15, K = 2*(L>>4)+{0,1} in {b.x,b.y}.
// C/D 16x16 layout: elem v of lane L -> (M = v + 8*(L>>4), N = L&15).
// ---------------------------------------------------------------------------
__global__ __launch_bounds__(32)
void proj_kernel(const float* __restrict__ F, const float* __restrict__ W,
                 const float* __restrict__ B, float* __restrict__ Z)
{
    int wave  = blockIdx.x;
    int c     = wave >> 8;         // 256 n-tiles per channel
    int ntile = wave & 255;
    int lane  = threadIdx.x;
    int l16   = lane & 15;
    int half  = lane >> 4;
    int row   = ntile * 16 + l16;

    const float* Fr = F + (size_t)row * IN_DIM;
    const float* Wc = W + (size_t)c * IN_DIM * D_DIM;

    v8f acc[4] = {};               // 4 d-tiles of 16 -> full D=64 per wave
    for (int k = 0; k < IN_DIM; k += 4) {
        int kk = k + 2 * half;
        v2f a;
        a.x = Fr[kk];
        a.y = Fr[kk + 1];
#pragma unroll
        for (int t = 0; t < 4; ++t) {
            v2f b;
            b.x = Wc[(size_t)kk       * D_DIM + t * 16 + l16];
            b.y = Wc[(size_t)(kk + 1) * D_DIM + t * 16 + l16];
            acc[t] = __builtin_amdgcn_wmma_f32_16x16x4_f32(
                false, a, false, b, (short)0, acc[t], false, false);
        }
    }

    // bias (depends only on d = t*16 + l16), then row-wise l2 norm over d=0..63
    float bv[4];
#pragma unroll
    for (int t = 0; t < 4; ++t) bv[t] = B[c * D_DIM + t * 16 + l16];

    float q[8];
#pragma unroll
    for (int v = 0; v < 8; ++v) {
        float s = 0.0f;
#pragma unroll
        for (int t = 0; t < 4; ++t) {
            float x = acc[t][v] + bv[t];
            acc[t][v] = x;
            s += x * x;
        }
        q[v] = s;
    }
    // reduce across the 16 lanes of each half-wave (rows live per-half)
#pragma unroll
    for (int v = 0; v < 8; ++v) {
        float s = q[v];
        s += __shfl_xor(s, 1, 32);
        s += __shfl_xor(s, 2, 32);
        s += __shfl_xor(s, 4, 32);
        s += __shfl_xor(s, 8, 32);
        q[v] = rsqrtf(fmaxf(s, 1e-12f));
    }
#pragma unroll
    for (int v = 0; v < 8; ++v) {
        int m = v + 8 * half;
        size_t base = ((size_t)c * N_NODES + ntile * 16 + m) * D_DIM;
        float r = q[v];
#pragma unroll
        for (int t = 0; t < 4; ++t)
            Z[base + t * 16 + l16] = acc[t][v] * r;
    }
}

// ---------------------------------------------------------------------------
// Kernel 2: one routing iteration, one 256-thread block (8 wave32) per node i.
//   scores s_c = <z[c,i,:], z[c,j,:]> for adj[i,j]>0; softmax over c;
//   z_out[c,i,:] = l2norm( z[c,i,:] + sum_j w_c(i,j) * z[c,j,:] )
// Edge processing: lane L owns (channel L>>2, d-segment (L&3)*16 .. +15).
// Deterministic: fixed j-ranges per wave, fixed-order LDS reduction, no atomics.
// ---------------------------------------------------------------------------
__global__ __launch_bounds__(256)
void route_kernel(const float* __restrict__ Zin, const float* __restrict__ Adj,
                  float* __restrict__ Zout)
{
    __shared__ float zi[C_CH * D_DIM];        //  2 KB: this node's z rows
    __shared__ float part[8 * C_CH * D_DIM];  // 16 KB: per-wave partial agg

    int i    = blockIdx.x;
    int tid  = threadIdx.x;
    int wave = tid >> 5;
    int lane = tid & 31;

    for (int idx = tid; idx < C_CH * D_DIM; idx += 256) {
        int c = idx >> 6, d = idx & 63;
        zi[idx] = Zin[((size_t)c * N_NODES + i) * D_DIM + d];
    }
    __syncthreads();

    int c    = lane >> 2;
    int dseg = (lane & 3) * 16;

    float riz[16], acc[16];
#pragma unroll
    for (int q = 0; q < 16; ++q) {
        riz[q] = zi[c * 64 + dseg + q];
        acc[q] = 0.0f;
    }

    const float* arow = Adj + (size_t)i * N_NODES;
    for (int it = 0; it < 16; ++it) {          // wave covers 512 columns
        int jbase = wave * 512 + it * 32;
        float a = arow[jbase + lane];
        unsigned long long mask = __ballot(a > 0.0f);
        while (mask) {
            int s = __ffsll(mask) - 1;
            mask &= mask - 1;
            int j = jbase + s;

            const float* zj = Zin + ((size_t)c * N_NODES + j) * D_DIM + dseg;
            float4 x0 = *(const float4*)(zj + 0);
            float4 x1 = *(const float4*)(zj + 4);
            float4 x2 = *(const float4*)(zj + 8);
            float4 x3 = *(const float4*)(zj + 12);
            float x[16] = { x0.x, x0.y, x0.z, x0.w, x1.x, x1.y, x1.z, x1.w,
                            x2.x, x2.y, x2.z, x2.w, x3.x, x3.y, x3.z, x3.w };

            float dp = 0.0f;
#pragma unroll
            for (int q = 0; q < 16; ++q) dp = fmaf(riz[q], x[q], dp);
            // reduce within 4-lane channel group -> all 4 lanes hold dot(c)
            dp += __shfl_xor(dp, 1, 32);
            dp += __shfl_xor(dp, 2, 32);

            // gather the 8 channel scores, softmax over channels (beta = 1)
            float dots[8];
#pragma unroll
            for (int cc = 0; cc < 8; ++cc) dots[cc] = __shfl(dp, cc * 4, 32);
            float mx = dots[0];
#pragma unroll
            for (int cc = 1; cc < 8; ++cc) mx = fmaxf(mx, dots[cc]);
            float esum = 0.0f, ec = 0.0f;
#pragma unroll
            for (int cc = 0; cc < 8; ++cc) {
                float e = __expf(dots[cc] - mx);
                esum += e;
                if (cc == c) ec = e;
            }
            float wgt = ec / esum;
#pragma unroll
            for (int q = 0; q < 16; ++q) acc[q] = fmaf(wgt, x[q], acc[q]);
        }
    }

#pragma unroll
    for (int q = 0; q < 16; ++q)
        part[wave * 512 + c * 64 + dseg + q] = acc[q];
    __syncthreads();

    // wave w reduces channel w; lane covers d = 2*lane, 2*lane+1
    {
        int cc = wave;
        int d0 = lane * 2, d1 = d0 + 1;
        float s0 = zi[cc * 64 + d0];
        float s1 = zi[cc * 64 + d1];
#pragma unroll
        for (int w = 0; w < 8; ++w) {
            s0 += part[w * 512 + cc * 64 + d0];
            s1 += part[w * 512 + cc * 64 + d1];
        }
        float ss = s0 * s0 + s1 * s1;
        ss += __shfl_xor(ss, 1, 32);
        ss += __shfl_xor(ss, 2, 32);
        ss += __shfl_xor(ss, 4, 32);
        ss += __shfl_xor(ss, 8, 32);
        ss += __shfl_xor(ss, 16, 32);
        float r = rsqrtf(fmaxf(ss, 1e-12f));
        size_t base = ((size_t)cc * N_NODES + i) * D_DIM;
        Zout[base + d0] = s0 * r;
        Zout[base + d1] = s1 * r;
    }
}

// ---------------------------------------------------------------------------
// Kernel 3: out[n,o] = h[n,:] @ W_o + bias,  h[n, c*64+d] = z[c,n,d]
// One wave per (16-row tile, 16-col tile). K = 512 in 128 f32 WMMA steps.
// ---------------------------------------------------------------------------
__global__ __launch_bounds__(32)
void out_kernel(const float* __restrict__ Z, const float* __restrict__ Wo,
                const float* __restrict__ Bias, float* __restrict__ Out)
{
    int wave  = blockIdx.x;        // 256 n-tiles * 8 o-tiles
    int ntile = wave >> 3;
    int otile = wave & 7;
    int lane  = threadIdx.x;
    int l16   = lane & 15;
    int half  = lane >> 4;
    int row   = ntile * 16 + l16;

    v8f acc = {};
    for (int k = 0; k < C_CH * D_DIM; k += 4) {
        int kk = k + 2 * half;
        int c = kk >> 6, d = kk & 63;         // kk even -> kk,kk+1 same channel
        const float* zr = Z + ((size_t)c * N_NODES + row) * D_DIM + d;
        v2f a;
        a.x = zr[0];
        a.y = zr[1];
        v2f b;
        b.x = Wo[(size_t)kk       * OUT_DIM + otile * 16 + l16];
        b.y = Wo[(size_t)(kk + 1) * OUT_DIM + otile * 16 + l16];
        acc = __builtin_amdgcn_wmma_f32_16x16x4_f32(
            false, a, false, b, (short)0, acc, false, false);
    }
    float bb = Bias[otile * 16 + l16];
#pragma unroll
    for (int v = 0; v < 8; ++v) {
        int m = v + 8 * half;
        Out[(size_t)(ntile * 16 + m) * OUT_DIM + otile * 16 + l16] = acc[v] + bb;
    }
}

// ---------------------------------------------------------------------------
extern "C" void kernel_launch(void* const* d_in, const int* in_sizes, int n_in,
                              void* d_out, int out_size, void* d_ws, size_t ws_size,
                              hipStream_t stream)
{
    const float* F    = (const float*)d_in[0];   // [4096, 256]
    const float* Adj  = (const float*)d_in[1];   // [4096, 4096]
    const float* W    = (const float*)d_in[2];   // [8, 256, 64]
    const float* B    = (const float*)d_in[3];   // [8, 1, 64]
    const float* Wo   = (const float*)d_in[4];   // [512, 128]
    const float* Bias = (const float*)d_in[5];   // [1, 128]
    float* Out = (float*)d_out;                  // [4096, 128]

    float* z0 = (float*)d_ws;                              // 8 MB
    float* z1 = z0 + (size_t)C_CH * N_NODES * D_DIM;       // 8 MB

    proj_kernel<<<C_CH * (N_NODES / 16), 32, 0, stream>>>(F, W, B, z0);

    float* zin = z0;
    float* zout = z1;
    for (int it = 0; it < ITERS; ++it) {
        route_kernel<<<N_NODES, 256, 0, stream>>>(zin, Adj, zout);
        float* tmp = zin; zin = zout; zout = tmp;
    }

    out_kernel<<<(N_NODES / 16) * (OUT_DIM / 16), 32, 0, stream>>>(zin, Wo, Bias, Out);
}